// SlaterDetSampler_ordered_78417512891184
// MI455X (gfx1250) — compile-verified
//
#include <hip/hip_runtime.h>
#include <math.h>

#define Dn 512
#define Nn 128

typedef __attribute__((ext_vector_type(2))) float v2f;
typedef __attribute__((ext_vector_type(8))) float v8f;

// D = A(16x4) * B(4x16) + C(16x16), fp32 WMMA (CDNA5 V_WMMA_F32_16X16X4_F32)
static __device__ __forceinline__ v8f wmma_f32_4(v2f a, v2f b, v8f c) {
#if defined(__AMDGCN__) && __has_builtin(__builtin_amdgcn_wmma_f32_16x16x4_f32)
  return __builtin_amdgcn_wmma_f32_16x16x4_f32(
      /*neg_a=*/false, a, /*neg_b=*/false, b,
      /*c_mod=*/(short)0, c, /*reuse_a=*/false, /*reuse_b=*/false);
#else
  (void)a; (void)b;
  return c;
#endif
}

// --------------------------------------------------------------------------
// G = I - P * P^T   (512x512, K=128), one 16x16 tile per wave via WMMA f32.
// Tile layout (32-bit 16x16 C/D): VGPR r -> row r (lanes 0-15) / r+8 (16-31),
// col = lane&15.  A (16x4): lane<16 -> K={0,1}, lane>=16 -> K={2,3}.
// --------------------------------------------------------------------------
__global__ void __launch_bounds__(256) build_g_kernel(const float* __restrict__ P,
                                                      float* __restrict__ G) {
  const int lane = threadIdx.x & 31;
  const int wid  = threadIdx.x >> 5;
  const int tile = blockIdx.x * 8 + wid;     // 0..1023  (32x32 tiles)
  const int bi = tile >> 5, bj = tile & 31;
  const int half = lane >> 4;                // 0 or 1
  const int l15  = lane & 15;

  const int m = bi * 16 + l15;               // A row
  const int c = bj * 16 + l15;               // B col (B = P^T slice)
  v8f acc = {};
  #pragma unroll 4
  for (int kk = 0; kk < 32; ++kk) {          // K = 128 in steps of 4
    const int k0 = 4 * kk + (half << 1);
    v2f a, b;
    a.x = P[m * Nn + k0];  a.y = P[m * Nn + k0 + 1];
    b.x = P[c * Nn + k0];  b.y = P[c * Nn + k0 + 1];
    acc = wmma_f32_4(a, b, acc);
  }
  const int col = bj * 16 + l15;
  #pragma unroll
  for (int r = 0; r < 8; ++r) {
    const int row = bi * 16 + r + (half << 3);
    G[row * Dn + col] = ((row == col) ? 1.0f : 0.0f) - acc[r];
  }
}

// --------------------------------------------------------------------------
// Blocked right-looking LU (NB=16, unpivoted) of A_k = G - diag(occ_prefix_k).
// One 256-thread workgroup per matrix; matrix lives in global (L2-resident,
// 1 MB each, 128 MB total < 192 MB L2).  Panel (L11/U11, U12, L21) staged in
// LDS; trailing update C -= L21*U12 as 16x16 WMMA f32 tiles over the 8 waves.
// Records diag(U) into piv[k][*].
// --------------------------------------------------------------------------
__global__ void __launch_bounds__(256) lu_kernel(const float* __restrict__ G,
                                                 const int* __restrict__ positions,
                                                 float* __restrict__ Abufs,
                                                 float* __restrict__ piv,
                                                 int kbase) {
  __shared__ float sdiag[16 * 17];    // 16x16 diag block (L11\U11), padded
  __shared__ float sU[16 * 512];      // U12 panel rows (stride 512)
  __shared__ float sL[496 * 16];      // L21 panel rows (stride 16)

  const int tid = threadIdx.x;
  const int k = kbase + blockIdx.x;
  float* A = Abufs + (size_t)blockIdx.x * (Dn * Dn);

  // Phase 0: A = G (streamed from L2-shared copy), then subtract occupied diag.
  for (int idx = tid; idx < Dn * Dn; idx += 256) A[idx] = G[idx];
  __syncthreads();
  for (int t = tid; t < k; t += 256) {
    const int p = positions[t];
    A[p * Dn + p] -= 1.0f;
  }
  __syncthreads();

  const int lane = tid & 31, wid = tid >> 5;
  const int half = lane >> 4, l15 = lane & 15;

  for (int j = 0; j < Dn; j += 16) {
    const int rem = Dn - j - 16;

    // ---- stage + factor 16x16 diagonal block (unpivoted LU) ----
    {
      const int r = tid >> 4, c = tid & 15;
      sdiag[r * 17 + c] = A[(j + r) * Dn + (j + c)];
    }
    __syncthreads();
    {
      const int r = tid >> 4, c = tid & 15;
      for (int t = 0; t < 16; ++t) {
        if (r > t && c == t) sdiag[r * 17 + t] /= sdiag[t * 17 + t];
        __syncthreads();
        if (r > t && c > t) sdiag[r * 17 + c] -= sdiag[r * 17 + t] * sdiag[t * 17 + c];
        __syncthreads();
      }
    }
    if (tid < 16) piv[k * Dn + j + tid] = sdiag[tid * 17 + tid];

    if (rem > 0) {
      // ---- U12 = L11^{-1} * A[j:j+16, j+16:]  (forward substitution) ----
      for (int cc = tid; cc < rem; cc += 256) {
        float x[16];
        #pragma unroll
        for (int t = 0; t < 16; ++t) x[t] = A[(j + t) * Dn + (j + 16 + cc)];
        #pragma unroll
        for (int t = 1; t < 16; ++t)
          for (int s = 0; s < t; ++s) x[t] -= sdiag[t * 17 + s] * x[s];
        #pragma unroll
        for (int t = 0; t < 16; ++t) sU[t * 512 + cc] = x[t];
      }
      // ---- L21 = A[j+16:, j:j+16] * U11^{-1}  (back substitution on rows) ----
      for (int rr = tid; rr < rem; rr += 256) {
        float x[16];
        #pragma unroll
        for (int t = 0; t < 16; ++t) x[t] = A[(j + 16 + rr) * Dn + (j + t)];
        #pragma unroll
        for (int t = 0; t < 16; ++t) {
          for (int s = 0; s < t; ++s) x[t] -= x[s] * sdiag[s * 17 + t];
          x[t] /= sdiag[t * 17 + t];
        }
        #pragma unroll
        for (int t = 0; t < 16; ++t) sL[rr * 16 + t] = x[t];
      }
    }
    __syncthreads();

    // ---- trailing update: A[j+16:, j+16:] -= L21 * U12 via WMMA f32 ----
    if (rem > 0) {
      const int nb = rem >> 4;
      const int ntiles = nb * nb;
      for (int tile = wid; tile < ntiles; tile += 8) {   // wave-uniform loop
        const int bi = tile / nb, bj = tile - bi * nb;
        v8f acc = {};
        #pragma unroll
        for (int kk = 0; kk < 4; ++kk) {                 // K = 16
          const int ak = 4 * kk + (half << 1);
          const int arow = bi * 16 + l15;
          const int bcol = bj * 16 + l15;
          v2f a, b;
          a.x = sL[arow * 16 + ak];      a.y = sL[arow * 16 + ak + 1];
          b.x = sU[ak * 512 + bcol];     b.y = sU[(ak + 1) * 512 + bcol];
          acc = wmma_f32_4(a, b, acc);
        }
        const int col  = j + 16 + bj * 16 + l15;
        const int row0 = j + 16 + bi * 16 + (half << 3);
        #pragma unroll
        for (int r = 0; r < 8; ++r) {
          float* p = &A[(row0 + r) * Dn + col];
          *p -= acc[r];                                   // L2-resident RMW
        }
      }
    }
    __syncthreads();
  }
}

// --------------------------------------------------------------------------
// probs[k,i] = exclusive-cumprod(q) * (1 - piv), masked + thresholded.
// One block per k; 512-thread Hillis-Steele product scan in LDS.
// --------------------------------------------------------------------------
__global__ void __launch_bounds__(512) probs_kernel(const float* __restrict__ piv,
                                                    const int* __restrict__ positions,
                                                    float* __restrict__ out) {
  __shared__ float q[512];
  const int k = blockIdx.x, i = threadIdx.x;
  const float p = piv[k * Dn + i];
  const int xmin = (k == 0) ? 0 : (positions[k - 1] + 1);
  const int xmax = Dn - Nn + k + 1;
  q[i] = (i >= xmin) ? p : 1.0f;
  __syncthreads();
  for (int off = 1; off < 512; off <<= 1) {
    const float v = (i >= off) ? q[i - off] : 1.0f;
    __syncthreads();
    q[i] *= v;
    __syncthreads();
  }
  const float c = (i == 0) ? 1.0f : q[i - 1];
  float pr = c * (1.0f - p);
  const bool valid = (i >= xmin) && (i < xmax);
  pr = valid ? pr : 0.0f;
  pr = (fabsf(pr) > 1e-15f) ? pr : 0.0f;
  out[k * Dn + i] = pr;
}

// log_prob_sample = sum_k log(probs[k, positions[k]])
__global__ void __launch_bounds__(128) logsum_kernel(const float* __restrict__ probs,
                                                     const int* __restrict__ positions,
                                                     float* __restrict__ out_scalar) {
  __shared__ float s[128];
  const int k = threadIdx.x;
  s[k] = logf(probs[k * Dn + positions[k]]);
  __syncthreads();
  for (int off = 64; off > 0; off >>= 1) {
    if (k < off) s[k] += s[k + off];
    __syncthreads();
  }
  if (k == 0) *out_scalar = s[0];
}

extern "C" void kernel_launch(void* const* d_in, const int* in_sizes, int n_in,
                              void* d_out, int out_size, void* d_ws, size_t ws_size,
                              hipStream_t stream) {
  (void)in_sizes; (void)n_in; (void)out_size;
  const float* P = (const float*)d_in[0];        // [512,128] f32
  const int* positions = (const int*)d_in[1];    // [128] i32
  float* out = (float*)d_out;                    // [128*512] probs + 1 scalar

  char* ws = (char*)d_ws;
  float* G     = (float*)ws;                                       // 1 MB
  float* piv   = (float*)(ws + (size_t)Dn * Dn * 4);               // 256 KB
  float* Abufs = (float*)(ws + (size_t)Dn * Dn * 4 + (size_t)Nn * Dn * 4);

  const size_t fixed = (size_t)Dn * Dn * 4 + (size_t)Nn * Dn * 4;
  const size_t per   = (size_t)Dn * Dn * 4;                        // 1 MB per matrix
  int slots = 1;
  if (ws_size > fixed + per) {
    size_t s = (ws_size - fixed) / per;
    slots = (int)(s > (size_t)Nn ? (size_t)Nn : s);
    if (slots < 1) slots = 1;
  }

  build_g_kernel<<<128, 256, 0, stream>>>(P, G);
  for (int kbase = 0; kbase < Nn; kbase += slots) {
    int batch = Nn - kbase;
    if (batch > slots) batch = slots;
    lu_kernel<<<batch, 256, 0, stream>>>(G, positions, Abufs, piv, kbase);
  }
  probs_kernel<<<Nn, 512, 0, stream>>>(piv, positions, out);
  logsum_kernel<<<1, 128, 0, stream>>>(out, positions, out + (size_t)Nn * Dn);
}